// PRSA_Net_34144990003589
// MI455X (gfx1250) — compile-verified
//
#include <hip/hip_runtime.h>
#include <hip/hip_bf16.h>

typedef float v2f __attribute__((ext_vector_type(2)));
typedef float v8f __attribute__((ext_vector_type(8)));

#define TT 200          // T_SCALE
#define BB 2            // batch
#define CH 256
#define FD 2048
#define DUR 64
#define RES 16
#define CCH 128         // cls-head channels
#define KCH 64          // K chunk staged in LDS for gemm_feat
#define LDSPAD 68       // padded k-stride (4*col mod 64 distinct -> conflict-free b64 reads)

// low 32 bits of a generic pointer to __shared__ = LDS byte offset
// (aperture: ADDR[63:32] = SHARED_BASE tag, ADDR[31:0] = offset)
__device__ __forceinline__ unsigned lds_off(const void* p) {
    return (unsigned)(unsigned long long)p;
}

__device__ __forceinline__ void async_load_b32(unsigned ldsoff, const void* gaddr) {
    asm volatile("global_load_async_to_lds_b32 %0, %1, off"
                 :: "v"(ldsoff), "v"((unsigned long long)gaddr) : "memory");
}
__device__ __forceinline__ void wait_async0() {
    asm volatile("s_wait_asynccnt 0" ::: "memory");
}

// ---------------------------------------------------------------------------
// Base conv1d: h[b,m,t] = relu(sum_{ic,tap} x[b,ic,t+tap-1]*W[m,ic,tap] + bias[m])
// GEMM M=256, K=2048x3taps, N=400.  grid (4, 25), block 128
// ---------------------------------------------------------------------------
__global__ void base_conv_wmma(const float* __restrict__ x,
                               const float* __restrict__ W,
                               const float* __restrict__ bias,
                               float* __restrict__ out) {
    const int lane  = threadIdx.x & 31;
    const int wave  = threadIdx.x >> 5;
    const int mBase = blockIdx.x * 64 + wave * 16;
    const int nBase = blockIdx.y * 16;
    const int row   = lane & 15;
    const int hi    = lane >> 4;
    const int n     = nBase + row;
    const int b     = n / TT;
    const int t     = n - b * TT;

    v8f acc = {};
    const float* Arow  = W + (long)(mBase + row) * (FD * 3);
    const float* Bbase = x + (long)b * (FD * TT);

    for (int tap = 0; tap < 3; ++tap) {
        const int tt = t + tap - 1;
        const bool ok = (tt >= 0) && (tt < TT);
        const float* Bp = Bbase + tt;           // + ic*TT
        const float* Ap = Arow + tap;           // + ic*3
#pragma unroll 4
        for (int k = 0; k < FD; k += 4) {
            const int ka = k + hi * 2;
            v2f a, bf;
            a.x  = Ap[(long)ka * 3];
            a.y  = Ap[(long)(ka + 1) * 3];
            bf.x = ok ? Bp[(long)ka * TT] : 0.0f;
            bf.y = ok ? Bp[(long)(ka + 1) * TT] : 0.0f;
            acc = __builtin_amdgcn_wmma_f32_16x16x4_f32(false, a, false, bf,
                                                        (short)0, acc, false, false);
        }
    }
#pragma unroll
    for (int i = 0; i < 8; ++i) {
        const int m = mBase + i + hi * 8;
        float v = acc[i] + bias[m];
        out[(long)b * (CH * TT) + (long)m * TT + t] = fmaxf(v, 0.0f);
    }
}

// ---------------------------------------------------------------------------
// Generic WMMA f32 GEMM, B operand [b][K][200], N = 400, K multiple of 64.
// B tile (16 n x 64 k) staged in LDS via async global->LDS loads, shared by
// all 4 waves (they differ only in M).  grid (M/64, 25, Z), block 128.
//   C[m,n] = sum_k A[m*ams + k*aks + z*azo] * Bf[b*K*200 + k*200 + t]
// epilogue: (+bias)(+relu)(+residual), out idx = b*obs + m*oms + t*ots + z*ozo
// ---------------------------------------------------------------------------
__global__ void gemm_feat_wmma(const float* __restrict__ A, int ams, int aks, long azo,
                               const float* __restrict__ Bf, int K,
                               const float* __restrict__ bias,
                               const float* __restrict__ resid,
                               float* __restrict__ out,
                               long obs, long oms, long ots, long ozo,
                               int do_relu) {
    __shared__ float Bt[16][LDSPAD];

    const int tid   = threadIdx.x;
    const int lane  = tid & 31;
    const int wave  = tid >> 5;
    const int mBase = blockIdx.x * 64 + wave * 16;
    const int nBase = blockIdx.y * 16;
    const int z     = blockIdx.z;
    A   += (long)z * azo;
    out += (long)z * ozo;

    const int row = lane & 15;
    const int hi  = lane >> 4;
    const int n   = nBase + row;
    const int b   = n / TT;
    const int t   = n - b * TT;

    const float* Ap = A + (long)(mBase + row) * ams;
    __builtin_prefetch(Ap, 0, 0);

    // fill metadata: this thread loads elements i = tid + e*128 of the 16x64 tile
    v8f acc = {};
    for (int k0 = 0; k0 < K; k0 += KCH) {
        // ---- async fill of B tile (whole block cooperates) ----
#pragma unroll
        for (int e = 0; e < (16 * KCH) / 128; ++e) {
            const int i  = tid + e * 128;
            const int ni = i >> 6;              // 0..15  (n within tile)
            const int kk = i & 63;              // 0..63  (k within chunk)
            const int nn = nBase + ni;
            const int bb = nn / TT;
            const int tt = nn - bb * TT;
            const float* gp = Bf + (long)bb * K * TT + (long)(k0 + kk) * TT + tt;
            async_load_b32(lds_off(&Bt[ni][kk]), gp);
        }
        wait_async0();
        __syncthreads();

        // ---- 16 WMMA steps from LDS ----
#pragma unroll
        for (int k = 0; k < KCH; k += 4) {
            const int kk = k + hi * 2;
            v2f a, bf;
            a.x  = Ap[(long)(k0 + kk) * aks];
            a.y  = Ap[(long)(k0 + kk + 1) * aks];
            bf.x = Bt[row][kk];
            bf.y = Bt[row][kk + 1];
            acc = __builtin_amdgcn_wmma_f32_16x16x4_f32(false, a, false, bf,
                                                        (short)0, acc, false, false);
        }
        __syncthreads();
    }
#pragma unroll
    for (int i = 0; i < 8; ++i) {
        const int m = mBase + i + hi * 8;
        float v = acc[i];
        if (bias) v += bias[m];
        if (do_relu) v = fmaxf(v, 0.0f);
        const long oidx = (long)b * obs + (long)m * oms + (long)t * ots;
        if (resid) v += resid[oidx];
        out[oidx] = v;
    }
}

// ---------------------------------------------------------------------------
// Banded attention: masks |t-s|<=4 and |t-s|<=8 -> softmax exactly zero
// outside band.  A_sum = softmax4 + softmax8 (full row written),
// attn_out = A_sum @ v.  grid (200, 2), block 32.
// ---------------------------------------------------------------------------
__global__ void banded_attn(const float* __restrict__ q, const float* __restrict__ k,
                            const float* __restrict__ v, const float* __restrict__ pos,
                            float* __restrict__ attn_out, float* __restrict__ A_out) {
    const int t = blockIdx.x, b = blockIdx.y, lane = threadIdx.x;
    const long fstride = (long)CH * TT;

    float sc[17];
#pragma unroll
    for (int j = 0; j < 17; ++j) sc[j] = 0.0f;

    const float* qp = q + (long)b * fstride + t;
    const float* kp = k + (long)b * fstride;
    for (int c = lane; c < CH; c += 32) {
        const float qv = qp[(long)c * TT];
        const float* kr = kp + (long)c * TT;
#pragma unroll
        for (int j = 0; j < 17; ++j) {
            const int s = t - 8 + j;
            if (s >= 0 && s < TT) sc[j] += qv * kr[s];
        }
    }
#pragma unroll
    for (int j = 0; j < 17; ++j)
        for (int off = 16; off > 0; off >>= 1)
            sc[j] += __shfl_xor(sc[j], off, 32);

    float m4 = -1e30f, m8 = -1e30f;
#pragma unroll
    for (int j = 0; j < 17; ++j) {
        const int s = t - 8 + j;
        const bool ok = (s >= 0) && (s < TT);
        const float sv = ok ? (sc[j] * 0.0625f + pos[(long)t * TT + s]) : -1e30f;
        sc[j] = sv;
        if (ok) {
            m8 = fmaxf(m8, sv);
            if (j >= 4 && j <= 12) m4 = fmaxf(m4, sv);
        }
    }
    float e4[17], e8[17], sum4 = 0.0f, sum8 = 0.0f;
#pragma unroll
    for (int j = 0; j < 17; ++j) {
        const int s = t - 8 + j;
        const bool ok = (s >= 0) && (s < TT);
        e8[j] = ok ? __expf(sc[j] - m8) : 0.0f;
        e4[j] = (ok && j >= 4 && j <= 12) ? __expf(sc[j] - m4) : 0.0f;
        sum8 += e8[j];
        sum4 += e4[j];
    }
    float Aj[17];
#pragma unroll
    for (int j = 0; j < 17; ++j) Aj[j] = e4[j] / sum4 + e8[j] / sum8;

    float* Arow = A_out + (long)b * (TT * TT) + (long)t * TT;
    for (int s = lane; s < TT; s += 32) {
        const int j = s - t + 8;
        Arow[s] = (j >= 0 && j <= 16) ? Aj[j] : 0.0f;
    }
    const float* vp = v + (long)b * fstride;
    for (int c = lane; c < CH; c += 32) {
        const float* vr = vp + (long)c * TT;
        float acc = 0.0f;
#pragma unroll
        for (int j = 0; j < 17; ++j) {
            const int s = t - 8 + j;
            if (s >= 0 && s < TT) acc += Aj[j] * vr[s];
        }
        attn_out[(long)b * fstride + (long)c * TT + t] = acc;
    }
}

// ---------------------------------------------------------------------------
// start/end heads: sigmoid(w . feat + b).  grid (200, 2, 2), block 32
// ---------------------------------------------------------------------------
__global__ void head_sigmoid(const float* __restrict__ sf, const float* __restrict__ ef,
                             const float* __restrict__ sw, const float* __restrict__ sb,
                             const float* __restrict__ ew, const float* __restrict__ eb,
                             float* __restrict__ os, float* __restrict__ oe) {
    const int t = blockIdx.x, b = blockIdx.y, which = blockIdx.z;
    const float* f = which ? ef : sf;
    const float* w = which ? ew : sw;
    float acc = 0.0f;
    for (int c = threadIdx.x; c < CH; c += 32)
        acc += f[(long)b * (CH * TT) + (long)c * TT + t] * w[c];
    for (int off = 16; off > 0; off >>= 1) acc += __shfl_xor(acc, off, 32);
    if (threadIdx.x == 0) {
        acc += which ? eb[0] : sb[0];
        float r = 1.0f / (1.0f + __expf(-acc));
        (which ? oe : os)[(long)b * TT + t] = r;
    }
}

// ---------------------------------------------------------------------------
// Fused align1d + 1x1 conv via precomputed P[b][r][t][oc].
// grid (200, 64, 2), block 128 (oc)
// ---------------------------------------------------------------------------
__global__ void align_fuse(const float* __restrict__ P, const float* __restrict__ cb1,
                           float* __restrict__ y1) {
    const int t = blockIdx.x, d = blockIdx.y, b = blockIdx.z;
    const int oc = threadIdx.x;
    const float s = (float)t, dur = (float)d;
    const bool va = (t + d) < TT;
    const float clen = dur + 1.0f;
    const float xmin = va ? (s - clen * 0.5f) : 0.0f;
    const float xmax = va ? (s + dur + clen * 0.5f) : 0.0f;

    float acc = cb1[oc];
    const float* Pb = P + (long)b * (RES * TT * CCH) + oc;
#pragma unroll
    for (int r = 0; r < RES; ++r) {
        const float frac = ((float)r + 0.5f) * (1.0f / RES);
        const float pos = xmin + (xmax - xmin) * frac;
        const float x0 = floorf(pos);
        const float w1 = pos - x0;
        const int x0i = (int)fminf(fmaxf(x0, 0.0f), (float)(TT - 1));
        const int x1i = (int)fminf(fmaxf(x0 + 1.0f, 0.0f), (float)(TT - 1));
        if (pos >= -1.0f && pos <= (float)TT) {
            const float* Pr = Pb + (long)r * (TT * CCH);
            acc += (1.0f - w1) * Pr[(long)x0i * CCH] + w1 * Pr[(long)x1i * CCH];
        }
    }
    y1[(((long)(b * CCH + oc)) * DUR + d) * TT + t] = fmaxf(acc, 0.0f);
}

// ---------------------------------------------------------------------------
// 3x3 conv 128->128 on 64x200 maps, implicit GEMM (M=128, K=128*9, N=25600).
// grid (2, 1600), block 128
// ---------------------------------------------------------------------------
__global__ void conv3x3_wmma(const float* __restrict__ in, const float* __restrict__ W,
                             const float* __restrict__ bias, float* __restrict__ out) {
    const int lane  = threadIdx.x & 31;
    const int wave  = threadIdx.x >> 5;
    const int mBase = blockIdx.x * 64 + wave * 16;
    const int nBase = blockIdx.y * 16;
    const int row = lane & 15;
    const int hi  = lane >> 4;
    const int n   = nBase + row;
    const int t   = n % TT;
    const int d   = (n / TT) % DUR;
    const int b   = n / (TT * DUR);

    v8f acc = {};
    const float* Arow = W + (long)(mBase + row) * (CCH * 9);
    __builtin_prefetch(Arow, 0, 0);
    const long plane = (long)DUR * TT;          // 12800
    for (int kd = 0; kd < 3; ++kd) {
        const int dd = d + kd - 1;
        for (int kt = 0; kt < 3; ++kt) {
            const int tt = t + kt - 1;
            const bool ok = (dd >= 0) && (dd < DUR) && (tt >= 0) && (tt < TT);
            const float* Bp = in + (long)b * CCH * plane + (long)dd * TT + tt;
            const float* Ap = Arow + kd * 3 + kt;
#pragma unroll 8
            for (int ic = 0; ic < CCH; ic += 4) {
                const int ka = ic + hi * 2;
                v2f a, bf;
                a.x  = Ap[(long)ka * 9];
                a.y  = Ap[(long)(ka + 1) * 9];
                bf.x = ok ? Bp[(long)ka * plane] : 0.0f;
                bf.y = ok ? Bp[(long)(ka + 1) * plane] : 0.0f;
                acc = __builtin_amdgcn_wmma_f32_16x16x4_f32(false, a, false, bf,
                                                            (short)0, acc, false, false);
            }
        }
    }
#pragma unroll
    for (int i = 0; i < 8; ++i) {
        const int m = mBase + i + hi * 8;
        const float v = fmaxf(acc[i] + bias[m], 0.0f);
        out[(((long)(b * CCH + m)) * DUR + d) * TT + t] = v;
    }
}

// ---------------------------------------------------------------------------
// Final 1x1 conv 128->2 + sigmoid.  thread per (b,d,t)
// ---------------------------------------------------------------------------
__global__ void conv4_sigmoid(const float* __restrict__ in, const float* __restrict__ W,
                              const float* __restrict__ bias, float* __restrict__ conf) {
    const int idx = blockIdx.x * blockDim.x + threadIdx.x;
    const int total = BB * DUR * TT;
    if (idx >= total) return;
    const int t = idx % TT;
    const int d = (idx / TT) % DUR;
    const int b = idx / (TT * DUR);
    const long plane = (long)DUR * TT;
    float a0 = bias[0], a1 = bias[1];
    const float* ip = in + (long)b * CCH * plane + (long)d * TT + t;
    for (int ic = 0; ic < CCH; ++ic) {
        const float v = ip[(long)ic * plane];
        a0 += v * W[ic];
        a1 += v * W[CCH + ic];
    }
    conf[(long)b * (2 * plane) + 0 * plane + (long)d * TT + t] = 1.0f / (1.0f + __expf(-a0));
    conf[(long)b * (2 * plane) + 1 * plane + (long)d * TT + t] = 1.0f / (1.0f + __expf(-a1));
}

// ---------------------------------------------------------------------------
extern "C" void kernel_launch(void* const* d_in, const int* in_sizes, int n_in,
                              void* d_out, int out_size, void* d_ws, size_t ws_size,
                              hipStream_t stream) {
    const float* x       = (const float*)d_in[0];
    const float* base_w  = (const float*)d_in[1];
    const float* base_b  = (const float*)d_in[2];
    const float* wq      = (const float*)d_in[3];
    const float* wk      = (const float*)d_in[4];
    const float* wv      = (const float*)d_in[5];
    const float* wo      = (const float*)d_in[6];
    const float* pos     = (const float*)d_in[7];
    const float* start_w = (const float*)d_in[8];
    const float* start_b = (const float*)d_in[9];
    const float* end_w   = (const float*)d_in[10];
    const float* end_b   = (const float*)d_in[11];
    const float* cw1     = (const float*)d_in[12];
    const float* cb1     = (const float*)d_in[13];
    const float* cw2     = (const float*)d_in[14];
    const float* cb2     = (const float*)d_in[15];
    const float* cw3     = (const float*)d_in[16];
    const float* cb3     = (const float*)d_in[17];
    const float* cw4     = (const float*)d_in[18];
    const float* cb4     = (const float*)d_in[19];

    float* out = (float*)d_out;
    float* o_conf  = out;                // conf   [2,2,64,200]
    float* o_start = out + 51200;        // start  [2,200]
    float* o_end   = out + 51600;        // end    [2,200]
    float* o_As    = out + 52000;        // A_s    [2,200,200]
    float* o_Ae    = out + 132000;
    float* o_Ap    = out + 212000;

    const long FEAT = (long)BB * CH * TT;        // 102400
    float* ws    = (float*)d_ws;
    float* h     = ws;
    float* basef = h     + FEAT;
    float* sf    = basef + FEAT;
    float* ef    = sf    + FEAT;
    float* cf    = ef    + FEAT;
    float* qb    = cf    + FEAT;
    float* kb    = qb    + FEAT;
    float* vb    = kb    + FEAT;
    float* attn  = vb    + FEAT;
    float* Adump = attn  + FEAT;                 // 80000 (layer-0 A, unused)
    float* P     = Adump + (long)BB * TT * TT;   // [b][r][t][oc] : 819200
    float* y1    = P     + (long)BB * RES * TT * CCH;
    float* y2    = y1    + (long)BB * CCH * DUR * TT;

    const dim3 blk128(128), blk32(32);

    base_conv_wmma<<<dim3(4, 25), blk128, 0, stream>>>(x, base_w, base_b, h);

    auto rslot = [&](const float* in_feat, int layer, float* out_feat, float* A_dest) {
        const long woff = (long)layer * CH * CH;
        gemm_feat_wmma<<<dim3(4, 25, 1), blk128, 0, stream>>>(
            wq + woff, CH, 1, 0, in_feat, CH, nullptr, nullptr,
            qb, (long)CH * TT, TT, 1, 0, 0);
        gemm_feat_wmma<<<dim3(4, 25, 1), blk128, 0, stream>>>(
            wk + woff, CH, 1, 0, in_feat, CH, nullptr, nullptr,
            kb, (long)CH * TT, TT, 1, 0, 0);
        gemm_feat_wmma<<<dim3(4, 25, 1), blk128, 0, stream>>>(
            wv + woff, CH, 1, 0, in_feat, CH, nullptr, nullptr,
            vb, (long)CH * TT, TT, 1, 0, 0);
        banded_attn<<<dim3(TT, BB), blk32, 0, stream>>>(qb, kb, vb, pos, attn, A_dest);
        gemm_feat_wmma<<<dim3(4, 25, 1), blk128, 0, stream>>>(
            wo + woff, CH, 1, 0, attn, CH, nullptr, in_feat,
            out_feat, (long)CH * TT, TT, 1, 0, 1);
    };

    rslot(h,     0, basef, Adump);
    rslot(basef, 1, sf,    o_As);
    rslot(basef, 2, ef,    o_Ae);
    head_sigmoid<<<dim3(TT, BB, 2), blk32, 0, stream>>>(sf, ef, start_w, start_b,
                                                        end_w, end_b, o_start, o_end);
    rslot(basef, 3, cf,    o_Ap);

    // P[b][r][t][oc] = sum_ch cw1[oc, ch*16+r] * cf[b,ch,t]   (grid.z = r)
    gemm_feat_wmma<<<dim3(2, 25, RES), blk128, 0, stream>>>(
        cw1, CH * RES, RES, 1, cf, CH, nullptr, nullptr,
        P, (long)RES * TT * CCH, 1, CCH, (long)TT * CCH, 0);

    align_fuse<<<dim3(TT, DUR, BB), blk128, 0, stream>>>(P, cb1, y1);

    conv3x3_wmma<<<dim3(2, 1600), blk128, 0, stream>>>(y1, cw2, cb2, y2);
    conv3x3_wmma<<<dim3(2, 1600), blk128, 0, stream>>>(y2, cw3, cb3, y1);

    conv4_sigmoid<<<dim3((BB * DUR * TT + 255) / 256), dim3(256), 0, stream>>>(
        y1, cw4, cb4, o_conf);
}